// eulerforward_12335146074256
// MI455X (gfx1250) — compile-verified
//
#include <hip/hip_runtime.h>

// Euler integration x_{i+1} = x_i + h*(W@x_i + b), H=1024, B=8192, 50 steps.
// fp32 GEMM on CDNA5 fp32 matrix pipe (v_wmma_f32_16x16x4_f32) with
// double-buffered LDS fed by gfx1250 async global->LDS copies (ASYNCcnt).
// X is staged in a K-pair-interleaved layout so B-fragments come out of one
// ds_load_2addr_b32 into an adjacent VGPR pair (no shuffle movs).

#define HIDDEN 1024
#define BATCH  8192
#define NSTEPS 50

typedef __attribute__((ext_vector_type(2))) float v2f;
typedef __attribute__((ext_vector_type(8))) float v8f;

#define BM 128               // block tile rows (W rows / output rows)
#define BN 256               // block tile cols (batch)
#define BK 32                // K-chunk staged in LDS
#define NCHUNK (HIDDEN / BK) // 32
#define WS_STRIDE (BK + 4)   // 36 floats: 16B-aligned rows, 8B-aligned even-k
#define XS_PITCH (2 * BN + 32) // 544 dwords per k-pair row; 544 % 64 == 32
                               // -> lanes 0-15 (pair r) and 16-31 (pair r+1)
                               //    land on disjoint bank halves

// Exact pointee type of the async builtin (AS1 global src, AS3 LDS dst).
typedef int v4i_vs __attribute__((vector_size(4 * sizeof(int))));
typedef __attribute__((address_space(1))) v4i_vs* gv4i_p;
typedef __attribute__((address_space(3))) v4i_vs* lv4i_p;

__device__ __forceinline__ void async_copy_b128(const void* gptr, void* lptr) {
#if __has_builtin(__builtin_amdgcn_global_load_async_to_lds_b128)
  __builtin_amdgcn_global_load_async_to_lds_b128(
      (gv4i_p)(unsigned long long)gptr, (lv4i_p)lptr, 0, 0);
#else
  unsigned loff = (unsigned)(unsigned long long)(lv4i_p)lptr;
  asm volatile("global_load_async_to_lds_b128 %0, %1, off"
               :: "v"(loff), "v"(gptr) : "memory");
#endif
}

__device__ __forceinline__ void wait_async0() {
#if __has_builtin(__builtin_amdgcn_s_wait_asynccnt)
  __builtin_amdgcn_s_wait_asynccnt(0);
#else
  asm volatile("s_wait_asynccnt 0x0" ::: "memory");
#endif
}

__global__ __launch_bounds__(256) void euler_step_wmma(
    const float* __restrict__ W, const float* __restrict__ xin,
    const float* __restrict__ bias, const float* __restrict__ t0p,
    const float* __restrict__ t1p, float* __restrict__ xout)
{
  __shared__ float Ws[2][BM * WS_STRIDE];        // W tile [M][K]
  __shared__ float Xs[2][(BK / 2) * XS_PITCH];   // x tile, k-pair interleaved

  const int tid   = threadIdx.x;
  const int lane  = tid & 31;
  const int wave  = tid >> 5;
  const int lhalf = lane >> 4;  // 0: lanes 0-15, 1: lanes 16-31
  const int l16   = lane & 15;

  const int m0 = blockIdx.y * BM;
  const int n0 = blockIdx.x * BN;

  const int wr = wave >> 2;  // 0..1 -> 64-row wave slice
  const int wc = wave & 3;   // 0..3 -> 64-col wave slice

  // cooperative async-copy coordinates (256 threads, b128 each)
  const int wtk = tid & 7;   // W: 8 x float4 per 32-float row
  const int wtm = tid >> 3;  // 32 rows per pass (4 passes)
  const int xtn = tid & 63;  // X: 64 x float4 per 256-float row
  const int xtk = tid >> 6;  // 4 rows per pass (8 passes)

  v8f acc[4][4];
#pragma unroll
  for (int i = 0; i < 4; ++i)
#pragma unroll
    for (int j = 0; j < 4; ++j)
      acc[i][j] = (v8f){0.f, 0.f, 0.f, 0.f, 0.f, 0.f, 0.f, 0.f};

  auto issue_chunk = [&](int buf, int k0) {
#pragma unroll
    for (int p = 0; p < 4; ++p)
      async_copy_b128(&W[(size_t)(m0 + wtm + 32 * p) * HIDDEN + k0 + 4 * wtk],
                      &Ws[buf][(wtm + 32 * p) * WS_STRIDE + 4 * wtk]);
#pragma unroll
    for (int p = 0; p < 8; ++p) {
      const int k = xtk + 4 * p;  // row within chunk, 0..31
      // interleaved dest: 16B block (k, xtn) -> (k/2)*PITCH + xtn*8 + (k&1)*4
      async_copy_b128(&xin[(size_t)(k0 + k) * BATCH + n0 + 4 * xtn],
                      &Xs[buf][(k >> 1) * XS_PITCH + 8 * xtn + 4 * (k & 1)]);
    }
  };

  issue_chunk(0, 0);

  for (int c = 0; c < NCHUNK; ++c) {
    const int cur = c & 1;
    wait_async0();    // this wave's copies into `cur` have landed
    __syncthreads();  // everyone's landed; everyone done reading `nxt`
    if (c + 1 < NCHUNK) issue_chunk(cur ^ 1, (c + 1) * BK);  // overlap w/ compute

#pragma unroll
    for (int kk = 0; kk < BK; kk += 4) {
      // A (16x4 f32): lanes 0-15 -> K=kk,kk+1 ; lanes 16-31 -> K=kk+2,kk+3
      v2f a[4];
#pragma unroll
      for (int mt = 0; mt < 4; ++mt) {
        const int m = 64 * wr + 16 * mt + l16;
        a[mt] = *(const v2f*)&Ws[cur][m * WS_STRIDE + kk + 2 * lhalf];
      }
      // B (4x16 f32): lane needs (kb, n) and (kb+1, n); in the interleaved
      // layout these are 4 dwords apart with a shared base -> one 2addr_b32.
      const int rbase = ((kk >> 1) + lhalf) * XS_PITCH + 128 * wc +
                        8 * (l16 >> 2) + (l16 & 3);
      v2f bf[4];
#pragma unroll
      for (int nt = 0; nt < 4; ++nt) {
        bf[nt].x = Xs[cur][rbase + 32 * nt];
        bf[nt].y = Xs[cur][rbase + 32 * nt + 4];
      }
#pragma unroll
      for (int mt = 0; mt < 4; ++mt)
#pragma unroll
        for (int nt = 0; nt < 4; ++nt)
          acc[mt][nt] = __builtin_amdgcn_wmma_f32_16x16x4_f32(
              false, a[mt], false, bf[nt], (short)0, acc[mt][nt], false, false);
    }
  }

  const float h = (t1p[0] - t0p[0]) * (1.0f / NSTEPS);

  // Epilogue: xout = xin + h*acc + h*bias.  C/D layout: VGPR r -> row r+8*lhalf.
#pragma unroll
  for (int mt = 0; mt < 4; ++mt) {
#pragma unroll
    for (int r = 0; r < 8; ++r) {
      const int row  = m0 + 64 * wr + 16 * mt + r + 8 * lhalf;
      const float hb = h * bias[row];
      const size_t rowbase = (size_t)row * BATCH;
#pragma unroll
      for (int nt = 0; nt < 4; ++nt) {
        const int col  = n0 + 64 * wc + 16 * nt + l16;
        const size_t i = rowbase + col;
        xout[i] = fmaf(h, acc[mt][nt][r], xin[i] + hb);
      }
    }
  }
}

extern "C" void kernel_launch(void* const* d_in, const int* in_sizes, int n_in,
                              void* d_out, int out_size, void* d_ws, size_t ws_size,
                              hipStream_t stream) {
  (void)in_sizes; (void)n_in; (void)out_size; (void)ws_size;
  // setup_inputs order: x0, t0, t1, W, b
  const float* x0 = (const float*)d_in[0];
  const float* t0 = (const float*)d_in[1];
  const float* t1 = (const float*)d_in[2];
  const float* W  = (const float*)d_in[3];
  const float* b  = (const float*)d_in[4];

  float* bufA = (float*)d_ws;   // 32 MB ping buffer
  float* bufB = (float*)d_out;  // pong buffer doubles as final output

  dim3 grid(BATCH / BN, HIDDEN / BM);  // (32, 8)
  dim3 block(256);

  for (int s = 0; s < NSTEPS; ++s) {
    const float* in = (s == 0) ? x0 : ((s & 1) ? bufA : bufB);
    float* out      = (s & 1) ? bufB : bufA;  // s=49 (last) -> bufB == d_out
    euler_step_wmma<<<grid, block, 0, stream>>>(W, in, b, t0, t1, out);
  }
}